// HOPNet_NoSequential_Layer_62483184222902
// MI455X (gfx1250) — compile-verified
//
#include <hip/hip_runtime.h>

typedef __attribute__((ext_vector_type(2))) float v2f;
typedef __attribute__((ext_vector_type(8))) float v8f;

// ---------------------------------------------------------------------------
// WMMA MLP kernel:  Y = relu( X @ W1 + b1 ) @ W2 + b2
// X = concat of NB blocks (each N x 64, row-major). W1 is (NB*64) x 64,
// W2 is 64 x 64.  One wave32 computes a 16x64 output tile with
// V_WMMA_F32_16X16X4_F32 (f32 exact, matching the f32 reference).
// NB is a template parameter so the block loop fully unrolls and every
// X pointer stays a kernel argument -> global_load_b64 (no FLAT fallback).
//
// f32 16x16x4 operand layouts (ISA 7.12.2):
//   A (16x4): lane L (L<16) holds row M=L, K={0,1}; lane L+16 holds K={2,3}
//   B (4x16): lane L holds col N=L&15, K pair selected by lane half
//   C/D     : VGPR i, lanes0-15 -> M=i, lanes16-31 -> M=i+8, N = lane&15
// ---------------------------------------------------------------------------
template <int NB>
__global__ __launch_bounds__(256) void mlp_wmma(
    const float* __restrict__ x0, const float* __restrict__ x1,
    const float* __restrict__ x2, const float* __restrict__ x3,
    const float* __restrict__ x4, const float* __restrict__ x5,
    const float* __restrict__ w1, const float* __restrict__ b1,
    const float* __restrict__ w2, const float* __restrict__ b2,
    float* __restrict__ y, int N)
{
  __shared__ float hid[8][16][66];   // per-wave 16x64 hidden tile (+pad, even)

  const int wave = threadIdx.x >> 5;
  const int lane = threadIdx.x & 31;
  const int mlo  = lane & 15;
  const int kh   = lane >> 4;              // selects K pair {0,1} vs {2,3}
  const int row0 = (blockIdx.x * 8 + wave) * 16;
  if (row0 >= N) return;                   // wave-uniform exit
  int r = row0 + mlo;                      // clamped load row (stores masked)
  if (r > N - 1) r = N - 1;

  // ---- layer 1: X @ W1 ----
  v8f acc[4] = {};
  #pragma unroll
  for (int kb = 0; kb < NB; ++kb) {
    const float* __restrict__ X =
        (kb == 0) ? x0 : (kb == 1) ? x1 : (kb == 2) ? x2 :
        (kb == 3) ? x3 : (kb == 4) ? x4 : x5;
    const float* __restrict__ Xr = X + (size_t)r * 64;
    const float* __restrict__ W  = w1 + (size_t)kb * 64 * 64;
    #pragma unroll 4
    for (int lk = 0; lk < 64; lk += 4) {
      const int kk = lk + kh * 2;
      v2f a = *(const v2f*)(Xr + kk);
      #pragma unroll
      for (int j = 0; j < 4; ++j) {
        const int n = j * 16 + mlo;
        v2f b;
        b.x = W[(size_t)kk * 64 + n];
        b.y = W[(size_t)(kk + 1) * 64 + n];
        acc[j] = __builtin_amdgcn_wmma_f32_16x16x4_f32(
            false, a, false, b, (short)0, acc[j], false, false);
      }
    }
  }

  // ---- bias + ReLU, re-stripe C-layout -> A-layout through LDS ----
  #pragma unroll
  for (int j = 0; j < 4; ++j) {
    const int n = j * 16 + mlo;
    const float bj = b1[n];
    #pragma unroll
    for (int i = 0; i < 8; ++i)
      hid[wave][i + kh * 8][n] = fmaxf(acc[j][i] + bj, 0.0f);
  }
  // hidden tile is produced and consumed by the same wave: DS-counter
  // ordering (s_wait_dscnt) is sufficient, no workgroup barrier needed.

  // ---- layer 2: hidden @ W2 ----
  v8f acc2[4] = {};
  #pragma unroll 4
  for (int lk = 0; lk < 64; lk += 4) {
    const int kk = lk + kh * 2;
    v2f a = *(const v2f*)(&hid[wave][mlo][kk]);
    #pragma unroll
    for (int j = 0; j < 4; ++j) {
      const int n = j * 16 + mlo;
      v2f b;
      b.x = w2[(size_t)kk * 64 + n];
      b.y = w2[(size_t)(kk + 1) * 64 + n];
      acc2[j] = __builtin_amdgcn_wmma_f32_16x16x4_f32(
          false, a, false, b, (short)0, acc2[j], false, false);
    }
  }

  #pragma unroll
  for (int j = 0; j < 4; ++j) {
    const int n = j * 16 + mlo;
    const float bj = b2[n];
    #pragma unroll
    for (int i = 0; i < 8; ++i) {
      const int row = row0 + i + kh * 8;
      if (row < N) y[(size_t)row * 64 + n] = acc2[j][i] + bj;
    }
  }
}

// ---------------------------------------------------------------------------
// Edge propagate: out[dst] += msg[src] * val   (64 lanes per edge, coalesced
// 256B gather + coalesced global_atomic_add_f32 scatter)
// ---------------------------------------------------------------------------
__global__ __launch_bounds__(256) void prop_scatter(
    const float* __restrict__ msg, const int* __restrict__ src,
    const int* __restrict__ dst, const float* __restrict__ val,
    float* __restrict__ out, int E)
{
  const long long t = (long long)blockIdx.x * blockDim.x + threadIdx.x;
  const int e = (int)(t >> 6);
  if (e >= E) return;
  const int c = (int)(t & 63);
  const int s = src[e];
  const int d = dst[e];
  const float v = val[e];
  atomicAdd(out + (size_t)d * 64 + c, msg[(size_t)s * 64 + c] * v);
}

__global__ __launch_bounds__(256) void count_edges(
    const int* __restrict__ dst, float* __restrict__ cnt, int E)
{
  const int e = blockIdx.x * blockDim.x + threadIdx.x;
  if (e >= E) return;
  atomicAdd(cnt + dst[e], 1.0f);
}

__global__ __launch_bounds__(256) void mean_div(
    float* __restrict__ buf, const float* __restrict__ cnt, int N)
{
  const long long t = (long long)blockIdx.x * blockDim.x + threadIdx.x;
  const int i = (int)(t >> 6);
  if (i >= N) return;
  const int c = (int)(t & 63);
  buf[(size_t)i * 64 + c] /= fmaxf(cnt[i], 1.0f);
}

// ---------------------------------------------------------------------------
extern "C" void kernel_launch(void* const* d_in, const int* in_sizes, int n_in,
                              void* d_out, int out_size, void* d_ws, size_t ws_size,
                              hipStream_t stream)
{
  const float* h0  = (const float*)d_in[0];
  const float* h1  = (const float*)d_in[1];
  const float* h2  = (const float*)d_in[2];
  const float* h3m = (const float*)d_in[3];
  const float* h3p = (const float*)d_in[4];
  const float* h4  = (const float*)d_in[5];

  const int N0 = in_sizes[0] / 64;
  const int N1 = in_sizes[1] / 64;
  const int N2 = in_sizes[2] / 64;
  const int N3 = in_sizes[3] / 64;
  const int N4 = in_sizes[5] / 64;
  const int E  = in_sizes[6];

  // adjacency accessors (inputs 6..41): name_src, name_dst, name_val
  enum { A010 = 0, A101, A232, B01, B02, B03, B04, B12, B13, B14, B23, B24 };
  auto SRC = [&](int a) { return (const int*)d_in[6 + 3 * a + 0]; };
  auto DST = [&](int a) { return (const int*)d_in[6 + 3 * a + 1]; };
  auto VAL = [&](int a) { return (const float*)d_in[6 + 3 * a + 2]; };

  // param accessors (inputs 42..145): per-MLP {w1,b1,w2,b2}
  // MSG_MLPS order: 0to0..0to4(0-4) 1to0..1to4(5-9) 2to0..2to4(10-14)
  //                 3to0..3to2(15-17) 4to0..4to2(18-20); u0..u4 = 21..25
  auto PW = [&](int p, int k) { return (const float*)d_in[42 + 4 * p + k]; };

  // ---- workspace layout (floats) ----
  float* W = (float*)d_ws;
  size_t o = 0;
  float* msgbuf = W + o; o += (size_t)N1 * 64;       // reused for every msg MLP
  float* zstart = W + o;                             // zero-init from here on
  float* m0to0  = W + o; o += (size_t)N0 * 64;
  float* m1to0  = W + o; o += (size_t)N0 * 64;
  float* m4to0  = W + o; o += (size_t)N0 * 64;
  float* m3to0s = W + o; o += (size_t)N0 * 64;       // plus+minus accumulated
  float* m0to1  = W + o; o += (size_t)N1 * 64;
  float* m1to1  = W + o; o += (size_t)N1 * 64;
  float* m4to1  = W + o; o += (size_t)N1 * 64;
  float* m3to1s = W + o; o += (size_t)N1 * 64;
  float* m0to2  = W + o; o += (size_t)N2 * 64;
  float* m1to2  = W + o; o += (size_t)N2 * 64;
  float* m2to2  = W + o; o += (size_t)N2 * 64;
  float* m3to2s = W + o; o += (size_t)N2 * 64;
  float* m4to2  = W + o; o += (size_t)N2 * 64;
  float* m0to3  = W + o; o += (size_t)N3 * 64;
  float* m1to3  = W + o; o += (size_t)N3 * 64;
  float* m2to3  = W + o; o += (size_t)N3 * 64;
  float* m0to4  = W + o; o += (size_t)N4 * 64;
  float* m1to4  = W + o; o += (size_t)N4 * 64;
  float* cnt04  = W + o; o += (size_t)N4;
  float* cnt14  = W + o; o += (size_t)N4;
  const size_t zbytes = (size_t)((W + o) - zstart) * sizeof(float);

  // ---- d_out layout: h0p h1p h2p h3p_minus h3p_plus h4p m2to0 m2to1 m2to4 ----
  float* out   = (float*)d_out;
  float* OH0   = out;
  float* OH1   = OH0 + (size_t)N0 * 64;
  float* OH2   = OH1 + (size_t)N1 * 64;
  float* OH3M  = OH2 + (size_t)N2 * 64;
  float* OH3P  = OH3M + (size_t)N3 * 64;
  float* OH4   = OH3P + (size_t)N3 * 64;
  float* OM20  = OH4 + (size_t)N4 * 64;              // m2to0 (also u0 input)
  float* OM21  = OM20 + (size_t)N0 * 64;             // m2to1 (also u1 input)
  float* OM24  = OM21 + (size_t)N1 * 64;             // m2to4 (also u4 input)

  // zero all scatter targets (graph-capture safe)
  hipMemsetAsync(zstart, 0, zbytes, stream);
  hipMemsetAsync(OM20, 0, (size_t)(N0 + N1 + N4) * 64 * sizeof(float), stream);

  // ---- launch helpers ----
  auto mlp1 = [&](int p, const float* x, float* y, int N) {
    mlp_wmma<1><<<dim3((N + 127) / 128), dim3(256), 0, stream>>>(
        x, nullptr, nullptr, nullptr, nullptr, nullptr,
        PW(p, 0), PW(p, 1), PW(p, 2), PW(p, 3), y, N);
  };
  auto mlp4 = [&](int p, const float* x0, const float* x1, const float* x2,
                  const float* x3, float* y, int N) {
    mlp_wmma<4><<<dim3((N + 127) / 128), dim3(256), 0, stream>>>(
        x0, x1, x2, x3, nullptr, nullptr,
        PW(p, 0), PW(p, 1), PW(p, 2), PW(p, 3), y, N);
  };
  auto mlp6 = [&](int p, const float* x0, const float* x1, const float* x2,
                  const float* x3, const float* x4, const float* x5,
                  float* y, int N) {
    mlp_wmma<6><<<dim3((N + 127) / 128), dim3(256), 0, stream>>>(
        x0, x1, x2, x3, x4, x5,
        PW(p, 0), PW(p, 1), PW(p, 2), PW(p, 3), y, N);
  };
  auto prop = [&](const float* msg, const int* s, const int* d, const float* v,
                  float* dstbuf) {
    const long long total = (long long)E * 64;
    prop_scatter<<<dim3((unsigned)((total + 255) / 256)), dim3(256), 0, stream>>>(
        msg, s, d, v, dstbuf, E);
  };

  // ---- message MLPs + propagations (stream order = dependency order) ----
  // h0 sources
  mlp1(0, h0, msgbuf, N0);  prop(msgbuf, SRC(A010), DST(A010), VAL(A010), m0to0);
  mlp1(1, h0, msgbuf, N0);  prop(msgbuf, SRC(B01),  DST(B01),  VAL(B01),  m0to1);
  mlp1(2, h0, msgbuf, N0);  prop(msgbuf, SRC(B02),  DST(B02),  VAL(B02),  m0to2);
  mlp1(3, h0, msgbuf, N0);  prop(msgbuf, SRC(B03),  DST(B03),  VAL(B03),  m0to3);
  mlp1(4, h0, msgbuf, N0);  prop(msgbuf, SRC(B04),  DST(B04),  VAL(B04),  m0to4);
  count_edges<<<dim3((E + 255) / 256), dim3(256), 0, stream>>>(DST(B04), cnt04, E);
  // h1 sources
  mlp1(5, h1, msgbuf, N1);  prop(msgbuf, DST(B01),  SRC(B01),  VAL(B01),  m1to0);
  mlp1(6, h1, msgbuf, N1);  prop(msgbuf, SRC(A101), DST(A101), VAL(A101), m1to1);
  mlp1(7, h1, msgbuf, N1);  prop(msgbuf, SRC(B12),  DST(B12),  VAL(B12),  m1to2);
  mlp1(8, h1, msgbuf, N1);  prop(msgbuf, SRC(B13),  DST(B13),  VAL(B13),  m1to3);
  mlp1(9, h1, msgbuf, N1);  prop(msgbuf, SRC(B14),  DST(B14),  VAL(B14),  m1to4);
  count_edges<<<dim3((E + 255) / 256), dim3(256), 0, stream>>>(DST(B14), cnt14, E);
  // h2 sources (m2to0 / m2to1 / m2to4 go straight to d_out)
  mlp1(10, h2, msgbuf, N2); prop(msgbuf, DST(B02),  SRC(B02),  VAL(B02),  OM20);
  mlp1(11, h2, msgbuf, N2); prop(msgbuf, DST(B12),  SRC(B12),  VAL(B12),  OM21);
  mlp1(12, h2, msgbuf, N2); prop(msgbuf, SRC(A232), DST(A232), VAL(A232), m2to2);
  mlp1(13, h2, msgbuf, N2); prop(msgbuf, SRC(B23),  DST(B23),  VAL(B23),  m2to3);
  mlp1(14, h2, msgbuf, N2); prop(msgbuf, SRC(B24),  DST(B24),  VAL(B24),  OM24);
  // h3 sources: plus and minus accumulate into the same buffer (only the
  // sum m3toX_plus + m3toX_minus is ever consumed)
  mlp1(15, h3p, msgbuf, N3); prop(msgbuf, DST(B03), SRC(B03), VAL(B03), m3to0s);
  mlp1(15, h3m, msgbuf, N3); prop(msgbuf, DST(B03), SRC(B03), VAL(B03), m3to0s);
  mlp1(16, h3p, msgbuf, N3); prop(msgbuf, DST(B13), SRC(B13), VAL(B13), m3to1s);
  mlp1(16, h3m, msgbuf, N3); prop(msgbuf, DST(B13), SRC(B13), VAL(B13), m3to1s);
  mlp1(17, h3p, msgbuf, N3); prop(msgbuf, DST(B23), SRC(B23), VAL(B23), m3to2s);
  mlp1(17, h3m, msgbuf, N3); prop(msgbuf, DST(B23), SRC(B23), VAL(B23), m3to2s);
  // h4 sources
  mlp1(18, h4, msgbuf, N4); prop(msgbuf, DST(B04), SRC(B04), VAL(B04), m4to0);
  mlp1(19, h4, msgbuf, N4); prop(msgbuf, DST(B14), SRC(B14), VAL(B14), m4to1);
  mlp1(20, h4, msgbuf, N4); prop(msgbuf, DST(B24), SRC(B24), VAL(B24), m4to2);

  // mean-normalize the two "mean" propagations
  {
    const long long t4 = (long long)N4 * 64;
    mean_div<<<dim3((unsigned)((t4 + 255) / 256)), dim3(256), 0, stream>>>(m0to4, cnt04, N4);
    mean_div<<<dim3((unsigned)((t4 + 255) / 256)), dim3(256), 0, stream>>>(m1to4, cnt14, N4);
  }

  // ---- update MLPs (concat handled as per-block GEMMs vs W1 row slices) ----
  mlp6(21, h0, m0to0, m1to0, OM20, m3to0s, m4to0, OH0, N0);
  mlp6(22, h1, m0to1, m1to1, OM21, m3to1s, m4to1, OH1, N1);
  mlp6(23, h2, m0to2, m1to2, m2to2, m3to2s, m4to2, OH2, N2);
  mlp4(24, h3m, m0to3, m1to3, m2to3, OH3M, N3);
  mlp4(24, h3p, m0to3, m1to3, m2to3, OH3P, N3);
  mlp4(25, h4, m0to4, m1to4, OM24, OH4, N4);
}